// GridSamplerNewTexture_81174881894726
// MI455X (gfx1250) — compile-verified
//
#include <hip/hip_runtime.h>
#include <hip/hip_bf16.h>

// grid_sample bilinear, align_corners=True, border padding.
// z:    [N=8, C=64, IH=256, IW=256]  f32
// grid: [N=8, H=512, W=512, 2]       f32 (normalized coords)
// out:  [N=8, C=64, H=512, W=512]    f32
//
// Memory-bound random-gather kernel:
//  - 1 thread per output pixel; weights + 4 clamped corner offsets computed
//    once, reused across all 64 channels.
//  - channel loop FULLY unrolled: channel plane strides become compile-time
//    constants that fold into the 24-bit instruction offset of saddr-form
//    global_load_b32 / global_store_b32 -> near-zero address VALU per access,
//    max memory-level parallelism (LOADcnt saturates at 63 in flight).
//  - z loads temporal: whole 128MB z fits the 192MB L2; coords are random so
//    every gather is an L2 hit -- keep it resident.
//  - grid load + out stores non-temporal (gfx1250 th:*_NT): 528MB streaming
//    traffic must not evict z.

typedef float v2f __attribute__((ext_vector_type(2)));

__global__ __launch_bounds__(256) void grid_sample_bilinear_kernel(
    const float* __restrict__ z,
    const float* __restrict__ grid,
    float* __restrict__ out)
{
    constexpr int IH = 256, IW = 256, C = 64, OH = 512, OW = 512;
    constexpr uint32_t IPLANE = IH * IW;   // 65536 elems per (n,c) z plane
    constexpr uint32_t OPLANE = OH * OW;   // 262144 elems per (n,c) out plane

    const int tid = blockIdx.x * blockDim.x + threadIdx.x;   // (n,h,w) flat
    const int w = tid & (OW - 1);
    const int h = (tid >> 9) & (OH - 1);

    // n is uniform across each 256-thread block (n = tid >> 18): make it
    // scalar so gather/store bases live in SGPR pairs (saddr addressing).
    const int n_u = __builtin_amdgcn_readfirstlane(tid >> 18);

    // Coalesced non-temporal 8B load of (gx, gy); grid flat layout matches tid.
    const v2f g = __builtin_nontemporal_load((const v2f*)grid + tid);

    // Unnormalize (align_corners=True)
    const float ix = (g.x + 1.0f) * 0.5f * (float)(IW - 1);
    const float iy = (g.y + 1.0f) * 0.5f * (float)(IH - 1);

    const float fx = floorf(ix);
    const float fy = floorf(iy);
    const float tx = ix - fx;
    const float ty = iy - fy;

    // Bilinear weights from UNclamped corners (matches reference)
    const float w_nw = (1.0f - tx) * (1.0f - ty);
    const float w_ne = tx * (1.0f - ty);
    const float w_sw = (1.0f - tx) * ty;
    const float w_se = tx * ty;

    // Border-clamped integer corner indices -- all 32-bit math
    const int ixi = (int)fx;
    const int iyi = (int)fy;
    const int x0 = min(max(ixi,     0), IW - 1);
    const int x1 = min(max(ixi + 1, 0), IW - 1);
    const int y0 = min(max(iyi,     0), IH - 1);
    const int y1 = min(max(iyi + 1, 0), IH - 1);

    // Per-lane u32 voffsets, computed ONCE, reused by all 64 channels.
    const uint32_t o00 = (uint32_t)(y0 * IW + x0);   // nw
    const uint32_t o01 = (uint32_t)(y0 * IW + x1);   // ne
    const uint32_t o10 = (uint32_t)(y1 * IW + x0);   // sw
    const uint32_t o11 = (uint32_t)(y1 * IW + x1);   // se
    const uint32_t obase = (uint32_t)(h * OW + w);

    // Wave-uniform 64-bit bases (SGPR pairs).
    const float* __restrict__ zb = z   + (size_t)n_u * (size_t)(C * IPLANE);
    float* __restrict__       ob = out + (size_t)n_u * (size_t)(C * OPLANE);

    // Full unroll: channel strides are literal immediates -> saddr loads with
    // constant offsets, no per-channel address arithmetic.
    #pragma unroll
    for (uint32_t c = 0; c < C; ++c) {
        const uint32_t cb = c * IPLANE;   // compile-time constant per copy
        const float v00 = zb[cb + o00];   // temporal: keep z hot in L2
        const float v01 = zb[cb + o01];
        const float v10 = zb[cb + o10];
        const float v11 = zb[cb + o11];
        const float r = fmaf(v00, w_nw,
                        fmaf(v01, w_ne,
                        fmaf(v10, w_sw, v11 * w_se)));
        // streaming 512MB output: non-temporal so it doesn't evict z
        __builtin_nontemporal_store(r, ob + (c * OPLANE + obase));
    }
}

extern "C" void kernel_launch(void* const* d_in, const int* in_sizes, int n_in,
                              void* d_out, int out_size, void* d_ws, size_t ws_size,
                              hipStream_t stream) {
    const float* z    = (const float*)d_in[0];   // 8*64*256*256
    const float* grid = (const float*)d_in[1];   // 8*512*512*2
    float* out        = (float*)d_out;           // 8*64*512*512

    constexpr int total_pixels = 8 * 512 * 512;  // 2,097,152 (exact multiple of 256)
    dim3 block(256);
    dim3 gridDim(total_pixels / 256);            // 8192 blocks = 65536 wave32
    grid_sample_bilinear_kernel<<<gridDim, block, 0, stream>>>(z, grid, out);
}